// VDP_first_Conv_31997506355856
// MI455X (gfx1250) — compile-verified
//
#include <hip/hip_runtime.h>
#include <math.h>

// ---------------------------------------------------------------------------
// VDP first conv layer for MI455X (gfx1250, wave32).
// Outputs (concatenated in d_out, fp32):
//   mu_out : (16,24,24,16)        = 147456 floats
//   Sigma  : (16,576,576,16)      = 84934656 floats  (~340 MB -> bandwidth bound)
// Strategy: fuse Gram(X X^T) + softplus-scale + diag-abs; Gram tiles via
// v_wmma_f32_16x16x4_f32 (K=25 padded to 28); non-temporal 128b stores.
// Softplus scales precomputed once into d_ws; diag-fix hoisted to a
// wave-uniform scalar branch (only 36/1296 tiles touch the diagonal).
// ---------------------------------------------------------------------------

typedef float v2f __attribute__((ext_vector_type(2)));
typedef float v4f __attribute__((ext_vector_type(4)));
typedef float v8f __attribute__((ext_vector_type(8)));

#define BATCH 16
#define IMG_H 28
#define IMG_W 28
#define OUT_HW 24
#define PDIM 576           // 24*24 patches
#define KNUM 16
#define KF 25              // 5*5 patch features
#define TILES 36           // 576 / 16 tiles per dim
#define WAVES_PER_WG 8

// ---------------- scale precompute: 16 softplus values ---------------------
__global__ void softplus_kernel(const float* __restrict__ w_sigma,
                                float* __restrict__ scales)
{
    int k = threadIdx.x;           // launched with exactly KNUM threads
    scales[k] = log1pf(expf(w_sigma[k]));
}

// ---------------- mu path: plain VALID conv, 1 in-channel, 16 out ----------
__global__ __launch_bounds__(256) void mu_conv_kernel(
    const float* __restrict__ img_all,   // [16,28,28]
    const float* __restrict__ w_mu,      // [5,5,1,16]
    float* __restrict__ mu_out)          // [16,24,24,16]
{
    int idx = blockIdx.x * 256 + threadIdx.x;      // exactly 147456 threads
    int co = idx & 15;
    int t  = idx >> 4;
    int ox = t % OUT_HW; t /= OUT_HW;
    int oy = t % OUT_HW;
    int b  = t / OUT_HW;
    const float* im = img_all + b * (IMG_H * IMG_W);
    float acc = 0.f;
#pragma unroll
    for (int ky = 0; ky < 5; ++ky)
#pragma unroll
        for (int kx = 0; kx < 5; ++kx)
            acc += im[(oy + ky) * IMG_W + (ox + kx)] * w_mu[(ky * 5 + kx) * KNUM + co];
    mu_out[idx] = acc;
}

// Expand one wave's 16x16 G tile into Sigma rows of 16 scaled channels.
// DIAG=true only for tq==0 tiles (q < 16 -> element k==q gets fabs).
template <bool DIAG>
__device__ __forceinline__ void expand_tile(
    const v8f& c, float* __restrict__ row_base,   // lane base: (p0, q) row
    const float scale[KNUM], int m)
{
#pragma unroll
    for (int v = 0; v < 8; ++v) {
        float g = c[v];
        if (!isfinite(g)) g = 0.f;                // NaN/Inf -> 0 (scale finite, <1)
        // dst = row_base + v * PDIM*KNUM floats (fits 24-bit store offset)
        float* dst = row_base + (size_t)v * (PDIM * KNUM);
        float vals[KNUM];
#pragma unroll
        for (int k = 0; k < KNUM; ++k) {
            float vv = scale[k] * g;
            if (DIAG && k == m) vv = fabsf(vv);   // q == tq*16+m == m when tq==0
            vals[k] = vv;
        }
#pragma unroll
        for (int i = 0; i < 4; ++i) {
            v4f o = { vals[4 * i + 0], vals[4 * i + 1],
                      vals[4 * i + 2], vals[4 * i + 3] };
            __builtin_nontemporal_store(o, (v4f*)dst + i);
        }
    }
}

// ---------------- Sigma path: WMMA Gram tile -> scaled broadcast -----------
__global__ __launch_bounds__(256) void sigma_gram_kernel(
    const float* __restrict__ img_all,   // [16,28,28]
    const float* __restrict__ scales,    // [16] precomputed softplus(w_sigma)
    float* __restrict__ sigma_out)       // [16,576,576,16]
{
    __shared__ float lds_img[IMG_H * IMG_W];       // 3136 B

    const int b = blockIdx.y;
    const float* img = img_all + b * (IMG_H * IMG_W);
    for (int i = threadIdx.x; i < IMG_H * IMG_W; i += 256)
        lds_img[i] = img[i];
    __syncthreads();

    const int lane = threadIdx.x & 31;
    const int wave = threadIdx.x >> 5;
    const int tile = blockIdx.x * WAVES_PER_WG + wave;   // 0..1295 exactly
    const int tp = tile / TILES;
    const int tq = tile % TILES;
    const int half = lane >> 4;      // which K-pair / which M-half for C
    const int m    = lane & 15;

    // per-channel softplus scales (uniform -> scalarized to s_load_b512)
    v4f s0 = *((const v4f*)scales + 0);
    v4f s1 = *((const v4f*)scales + 1);
    v4f s2 = *((const v4f*)scales + 2);
    v4f s3 = *((const v4f*)scales + 3);
    float scale[KNUM] = { s0.x, s0.y, s0.z, s0.w,  s1.x, s1.y, s1.z, s1.w,
                          s2.x, s2.y, s2.z, s2.w,  s3.x, s3.y, s3.z, s3.w };

    // A fragment rows (M) and B fragment rows (N) both come from X:
    //   X[row][k] = img[oy+ky][ox+kx], row = oy*24+ox, k = ky*5+kx
    const int rowA = tp * 16 + m;
    const int rowB = tq * 16 + m;
    const int oyA = rowA / OUT_HW, oxA = rowA % OUT_HW;
    const int oyB = rowB / OUT_HW, oxB = rowB % OUT_HW;

    v8f c = {};   // 16x16 f32 accumulator tile of G

#pragma unroll
    for (int kk = 0; kk < 7; ++kk) {               // K = 25 padded to 28
        v2f a, bb;
#pragma unroll
        for (int j = 0; j < 2; ++j) {
            int k  = kk * 4 + half * 2 + j;        // A-layout: VGPRj = K pair member
            int kc = (k < KF) ? k : 0;             // keep LDS index in-bounds
            int ky = kc / 5, kx = kc % 5;
            float av = lds_img[(oyA + ky) * IMG_W + (oxA + kx)];
            float bv = lds_img[(oyB + ky) * IMG_W + (oxB + kx)];
            if (k >= KF) { av = 0.f; bv = 0.f; }   // zero-pad, no divergence
            a[j] = av; bb[j] = bv;
        }
        // D = A(16x4) * B(4x16) + C ; full fp32
        c = __builtin_amdgcn_wmma_f32_16x16x4_f32(
                false, a, false, bb, (short)0, c, false, false);
    }

    // C layout: lane (half,m), VGPR v -> G[p = tp*16 + v+8*half][q = tq*16 + m]
    const int q  = tq * 16 + m;
    const int p0 = tp * 16 + half * 8;
    float* row_base = sigma_out
        + (size_t)b * PDIM * PDIM * KNUM
        + ((size_t)p0 * PDIM + q) * KNUM;

    if (tq == 0)   // wave-uniform scalar branch: diag tiles are 36/1296
        expand_tile<true >(c, row_base, scale, m);
    else
        expand_tile<false>(c, row_base, scale, m);
}

// ---------------------------------------------------------------------------
extern "C" void kernel_launch(void* const* d_in, const int* in_sizes, int n_in,
                              void* d_out, int out_size, void* d_ws, size_t ws_size,
                              hipStream_t stream)
{
    const float* mu_in   = (const float*)d_in[0];   // (16,28,28,1)
    const float* w_mu    = (const float*)d_in[1];   // (5,5,1,16)
    const float* w_sigma = (const float*)d_in[2];   // (16,)

    float* out    = (float*)d_out;
    float* mu_out = out;                                  // 147456 floats
    float* sigma  = out + BATCH * OUT_HW * OUT_HW * KNUM; // offset 64B-aligned
    float* scales = (float*)d_ws;                         // 16 floats scratch

    // softplus scales once (stream order guarantees visibility downstream)
    softplus_kernel<<<1, KNUM, 0, stream>>>(w_sigma, scales);

    // mu conv: 147456 elements, one per thread
    mu_conv_kernel<<<BATCH * OUT_HW * OUT_HW * KNUM / 256, 256, 0, stream>>>(
        mu_in, w_mu, mu_out);

    // Sigma: 1296 G-tiles per batch, 8 waves (tiles) per workgroup
    dim3 grid(TILES * TILES / WAVES_PER_WG, BATCH);
    sigma_gram_kernel<<<grid, 256, 0, stream>>>(mu_in, scales, sigma);
}